// DeltaVolumeDecoder_50328426774785
// MI455X (gfx1250) — compile-verified
//
#include <hip/hip_runtime.h>
#include <hip/hip_bf16.h>
#include <math.h>

// ---------------------------------------------------------------------------
// DeltaVolumeDecoder for MI455X (gfx1250, wave32).
// Memory-bound problem (~264 MB streamed -> ~11us floor @ 23.3 TB/s).
// Phase 1/3: coalesced float4 VALU streaming (WMMA f32 K=4 would waste 15/16
// of the tile on a GEMV). Phase 2: the real 8x8 matrix layers done with
// v_wmma_f32_16x16x4_f32 (two K=4 steps, accumulated in D).
// ---------------------------------------------------------------------------

typedef float v2f __attribute__((ext_vector_type(2)));
typedef float v8f __attribute__((ext_vector_type(8)));

#define VSIZE     128
#define NBUCKETS  64
// ws float layout: [0,512) = 64 buckets x 8 partial sums ; [512,520) = h[8]
#define WS_FLOATS (NBUCKETS * 8 + 8)

// ---------------------------------------------------------------- zero kernel
__global__ void k_zero(float* __restrict__ out, int n_out,
                       float* __restrict__ ws, int n_ws) {
  const int stride = gridDim.x * blockDim.x;
  for (int i = blockIdx.x * blockDim.x + threadIdx.x; i < n_out; i += stride)
    out[i] = 0.0f;
  for (int i = blockIdx.x * blockDim.x + threadIdx.x; i < n_ws; i += stride)
    ws[i] = 0.0f;
}

// ------------------------------------------------- phase 1: y8 = coords @ W0
// coords_flat[3i+t] = (inds[3i + (2-t)] - 64)/64.  W0 row-major (3N x 8).
// Each thread streams whole points (24 consecutive W0 floats = 6 float4,
// fully coalesced), keeps y8 in registers, wave-reduces via shfl_xor, then
// block-reduces through LDS and lands on one of 64 atomic buckets.
__global__ void k_phase1(const int* __restrict__ inds,
                         const float* __restrict__ W0,
                         float* __restrict__ buckets, int n) {
  float acc[8];
#pragma unroll
  for (int t = 0; t < 8; ++t) acc[t] = 0.0f;

  const int stride = gridDim.x * blockDim.x;
  for (int i = blockIdx.x * blockDim.x + threadIdx.x; i < n; i += stride) {
    const int i0 = inds[3 * i + 0];
    const int i1 = inds[3 * i + 1];
    const int i2 = inds[3 * i + 2];
    const float c0 = (float)(i2 - 64) * (1.0f / 64.0f);  // row 3i
    const float c1 = (float)(i1 - 64) * (1.0f / 64.0f);  // row 3i+1
    const float c2 = (float)(i0 - 64) * (1.0f / 64.0f);  // row 3i+2
    const float4* r = (const float4*)(W0 + (size_t)24 * (size_t)i);
    const float4 a0 = r[0], a1 = r[1];
    const float4 b0 = r[2], b1 = r[3];
    const float4 d0 = r[4], d1 = r[5];
    acc[0] = fmaf(c0, a0.x, fmaf(c1, b0.x, fmaf(c2, d0.x, acc[0])));
    acc[1] = fmaf(c0, a0.y, fmaf(c1, b0.y, fmaf(c2, d0.y, acc[1])));
    acc[2] = fmaf(c0, a0.z, fmaf(c1, b0.z, fmaf(c2, d0.z, acc[2])));
    acc[3] = fmaf(c0, a0.w, fmaf(c1, b0.w, fmaf(c2, d0.w, acc[3])));
    acc[4] = fmaf(c0, a1.x, fmaf(c1, b1.x, fmaf(c2, d1.x, acc[4])));
    acc[5] = fmaf(c0, a1.y, fmaf(c1, b1.y, fmaf(c2, d1.y, acc[5])));
    acc[6] = fmaf(c0, a1.z, fmaf(c1, b1.z, fmaf(c2, d1.z, acc[6])));
    acc[7] = fmaf(c0, a1.w, fmaf(c1, b1.w, fmaf(c2, d1.w, acc[7])));
  }

  // wave32 butterfly reduction
#pragma unroll
  for (int off = 16; off >= 1; off >>= 1) {
#pragma unroll
    for (int t = 0; t < 8; ++t)
      acc[t] += __shfl_xor(acc[t], off, 32);
  }

  __shared__ float red[8][8];  // [wave][component], 256 threads = 8 waves
  const int lane = threadIdx.x & 31;
  const int wav  = threadIdx.x >> 5;
  if (lane == 0) {
#pragma unroll
    for (int t = 0; t < 8; ++t) red[wav][t] = acc[t];
  }
  __syncthreads();
  if (threadIdx.x < 8) {
    float s = 0.0f;
#pragma unroll
    for (int w = 0; w < 8; ++w) s += red[w][threadIdx.x];
    atomicAdd(&buckets[(blockIdx.x & (NBUCKETS - 1)) * 8 + threadIdx.x], s);
  }
}

// ----------------------------------- phase 2: sin MLP on the single 8-vector
// Single wave32.  Each 8x8 layer = two accumulated v_wmma_f32_16x16x4_f32.
// A (16x4, A[m,k]=x[k] replicated over M): lanes<16 V0=x[k0],V1=x[k0+1];
//                                          lanes>=16 V0=x[k0+2],V1=x[k0+3].
// B (4x16): lanes<16 V0=B[k0+0,n],V1=B[k0+1,n]; lanes>=16 rows k0+2/k0+3.
// D (16x16): VGPR0 lane n holds D[0,n] -> row 0 is the result vector.
__global__ void k_phase2(const float* __restrict__ buckets,
                         const float* __restrict__ b0,
                         const float* __restrict__ W1, const float* __restrict__ b1,
                         const float* __restrict__ W2, const float* __restrict__ b2,
                         const float* __restrict__ W3, const float* __restrict__ b3,
                         const float* __restrict__ W4, const float* __restrict__ b4,
                         float* __restrict__ hout) {
  const int lane = threadIdx.x;          // 0..31, uniform launch of 32
  const int n8   = lane & 7;
  const int nn   = lane & 15;
  const int hi   = (lane >> 4) & 1;      // 0: lanes 0-15, 1: lanes 16-31

  // fold the 64 bucket partials (every lane computes its &7 component: no
  // divergence, EXEC stays all-ones for the WMMAs below)
  float y = 0.0f;
  for (int b = 0; b < NBUCKETS; ++b) y += buckets[b * 8 + n8];

  // x = sin(coords@W0 + b0) ; lane L carries x[L&7]
  float xl = __builtin_sinf(y + b0[n8]);

  const float* Ws[4] = {W1, W2, W3, W4};
  const float* bs[4] = {b1, b2, b3, b4};

  for (int l = 0; l < 4; ++l) {
    const float* W = Ws[l];
    v8f d = {};
#pragma unroll
    for (int k0 = 0; k0 < 8; k0 += 4) {
      const int ka = k0 + (hi ? 2 : 0);
      const int kb = k0 + (hi ? 3 : 1);
      v2f A, B;
      A.x = __shfl(xl, ka, 32);          // x[ka]
      A.y = __shfl(xl, kb, 32);          // x[kb]
      const float wa = W[ka * 8 + (nn & 7)];
      const float wb = W[kb * 8 + (nn & 7)];
      B.x = (nn < 8) ? wa : 0.0f;        // cols 8..15 padded with zeros
      B.y = (nn < 8) ? wb : 0.0f;
      d = __builtin_amdgcn_wmma_f32_16x16x4_f32(
              /*neg_a=*/false, A, /*neg_b=*/false, B,
              /*c_mod=*/(short)0, d, /*reuse_a=*/false, /*reuse_b=*/false);
    }
    const float v = __shfl(d[0], n8, 32);   // D[0, n8] = (x @ W)[n8]
    xl = xl + __builtin_sinf(v + bs[l][n8]);
  }

  if (lane < 8) hout[lane] = xl;
}

// ------------------- phase 3: per-point W5 decode + trilinear atomic scatter
__global__ void k_phase3(const int* __restrict__ inds,
                         const float* __restrict__ refv,
                         const float* __restrict__ W5,
                         const float* __restrict__ b5,
                         const float* __restrict__ h8,
                         float* __restrict__ vol, int n) {
  const int i = blockIdx.x * blockDim.x + threadIdx.x;
  if (i >= n) return;

  float h[8];
#pragma unroll
  for (int k = 0; k < 8; ++k) h[k] = h8[k];   // uniform -> scalar-cached

  const size_t rowstride = 4 * (size_t)n;     // W5 is (8 x 4N) row-major
  float4 o = *(const float4*)(b5 + 4 * (size_t)i);
#pragma unroll
  for (int k = 0; k < 8; ++k) {
    const float4 w = *(const float4*)(W5 + k * rowstride + 4 * (size_t)i);
    o.x = fmaf(h[k], w.x, o.x);
    o.y = fmaf(h[k], w.y, o.y);
    o.z = fmaf(h[k], w.z, o.z);
    o.w = fmaf(h[k], w.w, o.w);
  }

  const int i0 = inds[3 * i + 0];
  const int i1 = inds[3 * i + 1];
  const int i2 = inds[3 * i + 2];
  const float cn0 = (float)(i2 - 64) * (1.0f / 64.0f);
  const float cn1 = (float)(i1 - 64) * (1.0f / 64.0f);
  const float cn2 = (float)(i0 - 64) * (1.0f / 64.0f);

  const float cx = 64.0f * (cn0 + o.x) + 64.0f;
  const float cy = 64.0f * (cn1 + o.y) + 64.0f;
  const float cz = 64.0f * (cn2 + o.z) + 64.0f;

  float v = refv[i] + o.w;
  v = v > 0.0f ? v : 0.0f;                    // relu

  const float bx = floorf(cx), by = floorf(cy), bz = floorf(cz);
  const int px = (int)bx, py = (int)by, pz = (int)bz;
  const float fx = cx - bx, fy = cy - by, fz = cz - bz;
  const float gx = 1.0f - fx, gy = 1.0f - fy, gz = 1.0f - fz;

#pragma unroll
  for (int oz = 0; oz < 2; ++oz)
#pragma unroll
    for (int oy = 0; oy < 2; ++oy)
#pragma unroll
      for (int ox = 0; ox < 2; ++ox) {
        const float amp = v * (ox ? fx : gx) * (oy ? fy : gy) * (oz ? fz : gz);
        // JAX .at[].add semantics: wrap negative indices, drop >= V
        int X = px + ox; if (X < 0) X += VSIZE;
        int Y = py + oy; if (Y < 0) Y += VSIZE;
        int Z = pz + oz; if (Z < 0) Z += VSIZE;
        if ((unsigned)X < VSIZE && (unsigned)Y < VSIZE && (unsigned)Z < VSIZE)
          atomicAdd(&vol[((size_t)Z * VSIZE + Y) * VSIZE + X], amp);
      }
}

// ---------------------------------------------------------------------------
extern "C" void kernel_launch(void* const* d_in, const int* in_sizes, int n_in,
                              void* d_out, int out_size, void* d_ws, size_t ws_size,
                              hipStream_t stream) {
  const int*   inds = (const int*)  d_in[0];
  const float* refv = (const float*)d_in[1];
  const float* W0   = (const float*)d_in[2];
  const float* b0   = (const float*)d_in[3];
  const float* W1   = (const float*)d_in[4];
  const float* b1   = (const float*)d_in[5];
  const float* W2   = (const float*)d_in[6];
  const float* b2   = (const float*)d_in[7];
  const float* W3   = (const float*)d_in[8];
  const float* b3   = (const float*)d_in[9];
  const float* W4   = (const float*)d_in[10];
  const float* b4   = (const float*)d_in[11];
  const float* W5   = (const float*)d_in[12];
  const float* b5   = (const float*)d_in[13];

  const int n = in_sizes[0] / 3;             // number of points
  float* ws      = (float*)d_ws;
  float* buckets = ws;                       // 64 x 8
  float* hvec    = ws + NBUCKETS * 8;        // 8

  k_zero<<<2048, 256, 0, stream>>>((float*)d_out, out_size, ws, WS_FLOATS);
  k_phase1<<<1024, 256, 0, stream>>>(inds, W0, buckets, n);
  k_phase2<<<1, 32, 0, stream>>>(buckets, b0, W1, b1, W2, b2, W3, b3, W4, b4, hvec);
  k_phase3<<<(n + 255) / 256, 256, 0, stream>>>(inds, refv, W5, b5, hvec,
                                                (float*)d_out, n);
}